// GCNResnet_36867999269282
// MI455X (gfx1250) — compile-verified
//
#include <hip/hip_runtime.h>

typedef __attribute__((ext_vector_type(2))) float v2f;
typedef __attribute__((ext_vector_type(8))) float v8f;

#define NEG_SLOPE 0.01f

// ---------------- degree / norm ----------------

__global__ void k_deg_init(float* __restrict__ deg, int N) {
    int i = blockIdx.x * blockDim.x + threadIdx.x;
    if (i < N) deg[i] = 1.0f;  // self-loop contributes 1 to every node's degree
}

__global__ void k_deg_count(const long long* __restrict__ ei, float* __restrict__ deg,
                            long long E) {
    long long e = (long long)blockIdx.x * blockDim.x + threadIdx.x;
    if (e < E) atomicAdd(&deg[(int)ei[E + e]], 1.0f);
}

__global__ void k_dinv(const float* __restrict__ deg, float* __restrict__ dinv, int N) {
    int i = blockIdx.x * blockDim.x + threadIdx.x;
    if (i < N) dinv[i] = rsqrtf(deg[i]);  // deg >= 1 always (self-loops)
}

// ---------------- fp32 WMMA GEMM: C[M,NOUT] = A[M,K] @ B[K,NOUT] ----------------
// One wave32 per 16x16 output tile; K-loop of V_WMMA_F32_16X16X4_F32.
// VGPR layouts (ISA 7.12.2):
//   A 16x4 f32 : lanes 0-15 -> M=lane, v0=K0,v1=K1 ; lanes 16-31 -> M=lane-16, v0=K2,v1=K3
//   B 4x16 f32 : lanes 0-15 -> N=lane, v0=K0,v1=K1 ; lanes 16-31 -> N=lane-16, v0=K2,v1=K3
//   C/D 16x16  : VGPR r -> M=r (lanes 0-15) / M=r+8 (lanes 16-31), N=lane&15

template <int K, int NOUT>
__global__ __launch_bounds__(256) void k_gemm_wmma(const float* __restrict__ A,
                                                   const float* __restrict__ B,
                                                   float* __restrict__ C, int M) {
    const int tilesN = NOUT / 16;
    int wave  = blockIdx.x * (blockDim.x >> 5) + (threadIdx.x >> 5);
    int tiles = ((M + 15) >> 4) * tilesN;
    if (wave >= tiles) return;  // wave-uniform: EXEC stays all-1s for WMMA

    int tm = wave / tilesN;
    int tn = wave % tilesN;
    int lane = threadIdx.x & 31;
    int m    = lane & 15;
    int half = lane >> 4;

    int rowA = tm * 16 + m;
    if (rowA >= M) rowA = M - 1;  // clamp loads for ragged tail; stores guarded below

    const float* ap = A + (size_t)rowA * K + half * 2;
    const float* bp = B + (size_t)(half * 2) * NOUT + tn * 16 + m;

    v8f acc = {};
#pragma unroll
    for (int k0 = 0; k0 < K; k0 += 4) {
        v2f a;
        a.x = ap[0];
        a.y = ap[1];
        v2f b;
        b.x = bp[0];
        b.y = bp[NOUT];
        acc = __builtin_amdgcn_wmma_f32_16x16x4_f32(
            /*neg_a=*/false, a, /*neg_b=*/false, b,
            /*c_mod=*/(short)0, acc, /*reuse_a=*/false, /*reuse_b=*/false);
        ap += 4;
        bp += 4 * NOUT;
    }

    int col   = tn * 16 + m;
    int rbase = tm * 16 + half * 8;
#pragma unroll
    for (int r = 0; r < 8; ++r) {
        int row = rbase + r;
        if (row < M) C[(size_t)row * NOUT + col] = acc[r];
    }
}

// ---------------- self-loop init: y = dinv[i]^2 * h + bias ----------------

template <int F>
__global__ void k_init_self(const float* __restrict__ h, const float* __restrict__ dinv,
                            const float* __restrict__ bias, float* __restrict__ y, int N) {
    long long t = (long long)blockIdx.x * blockDim.x + threadIdx.x;
    if (t >= (long long)N * F) return;
    int i = (int)(t / F);
    int f = (int)(t % F);
    float dv = dinv[i];
    y[t] = dv * dv * h[t] + bias[f];
}

// ---------------- edge scatter: y[dst] += norm * h[src]  (wave per edge) ----------------

template <int F, int VEC>
__global__ void k_scatter(const long long* __restrict__ ei, const float* __restrict__ dinv,
                          const float* __restrict__ h, float* __restrict__ y, long long E) {
    const int LPE = F / VEC;  // lanes per edge (== 32)
    long long t = (long long)blockIdx.x * blockDim.x + threadIdx.x;
    long long e = t / LPE;
    if (e >= E) return;
    int l = (int)(t % LPE);
    int s = (int)ei[e];
    int d = (int)ei[E + e];
    float nrm = dinv[s] * dinv[d];
    const float* hp = h + (size_t)s * F + l * VEC;
    float* yp = y + (size_t)d * F + l * VEC;
#pragma unroll
    for (int i = 0; i < VEC; ++i) atomicAdd(yp + i, hp[i] * nrm);
}

// ---------------- leaky relu / residual ----------------

__global__ void k_lrelu(float* __restrict__ y, long long n) {
    long long t = (long long)blockIdx.x * blockDim.x + threadIdx.x;
    if (t < n) {
        float v = y[t];
        y[t] = (v >= 0.0f) ? v : NEG_SLOPE * v;
    }
}

__global__ void k_residual(const float* __restrict__ x, const float* __restrict__ y,
                           float* __restrict__ out, long long n) {
    long long t = (long long)blockIdx.x * blockDim.x + threadIdx.x;
    if (t < n) out[t] = x[t] + y[t];
}

// ---------------- launch ----------------

extern "C" void kernel_launch(void* const* d_in, const int* in_sizes, int n_in,
                              void* d_out, int out_size, void* d_ws, size_t ws_size,
                              hipStream_t stream) {
    const float* x       = (const float*)d_in[0];
    const float* W1      = (const float*)d_in[1];
    const float* b1      = (const float*)d_in[2];
    const float* W2      = (const float*)d_in[3];
    const float* b2      = (const float*)d_in[4];
    const long long* ei  = (const long long*)d_in[5];

    const int NF = in_sizes[4];          // 64
    const int NH = in_sizes[2];          // 128
    const int N  = in_sizes[0] / NF;     // 50000
    const long long E = in_sizes[5] / 2; // 800000

    float* ws   = (float*)d_ws;
    float* deg  = ws;                       // N
    float* dinv = deg + N;                  // N
    float* h    = dinv + N;                 // N*NH (layer-1 h; reused for layer-2 h)
    float* y1   = h + (size_t)N * NH;       // N*NH
    float* y2   = y1 + (size_t)N * NH;      // N*NF

    const int T = 256;
    auto blocks = [](long long work, int t) { return (int)((work + t - 1) / t); };

    // symmetric normalization
    k_deg_init<<<blocks(N, T), T, 0, stream>>>(deg, N);
    k_deg_count<<<blocks(E, T), T, 0, stream>>>(ei, deg, E);
    k_dinv<<<blocks(N, T), T, 0, stream>>>(deg, dinv, N);

    // layer 1: h = x @ W1 ; y1 = aggregate + b1 ; leaky_relu
    {
        int tiles = ((N + 15) / 16) * (128 / 16);
        k_gemm_wmma<64, 128><<<(tiles + 7) / 8, T, 0, stream>>>(x, W1, h, N);
        k_init_self<128><<<blocks((long long)N * 128, T), T, 0, stream>>>(h, dinv, b1, y1, N);
        k_scatter<128, 4><<<blocks(E * 32, T), T, 0, stream>>>(ei, dinv, h, y1, E);
        k_lrelu<<<blocks((long long)N * 128, T), T, 0, stream>>>(y1, (long long)N * 128);
    }

    // layer 2: h2 = y1 @ W2 ; y2 = aggregate + b2 ; out = x + y2
    {
        int tiles = ((N + 15) / 16) * (64 / 16);
        k_gemm_wmma<128, 64><<<(tiles + 7) / 8, T, 0, stream>>>(y1, W2, h, N);
        k_init_self<64><<<blocks((long long)N * 64, T), T, 0, stream>>>(h, dinv, b2, y2, N);
        k_scatter<64, 2><<<blocks(E * 32, T), T, 0, stream>>>(ei, dinv, h, y2, E);
        k_residual<<<blocks((long long)N * 64, T), T, 0, stream>>>(x, y2, (float*)d_out,
                                                                   (long long)N * 64);
    }
}